// CoeffToValue_71554155151984
// MI455X (gfx1250) — compile-verified
//
#include <hip/hip_runtime.h>
#include <cstdint>

typedef __attribute__((ext_vector_type(2))) float v2f;
typedef __attribute__((ext_vector_type(8))) float v8f;

// Problem constants (coeff: (2,2,160,192,160) fp32, row-major)
#define DD 160
#define HH 192
#define WW 160
#define TDH 16                         // output tile height (h)
#define TDW 16                         // output tile width  (w)
#define DBLK 8                         // d-slices per block (one per wave)
#define NSLICE (DBLK + 2)              // 10 staged d-slices (with halo)
#define NROW (TDH + 2)                 // 18 staged rows (with halo)
#define NCOL (TDW + 2)                 // 18 staged cols (with halo)
#define LROW 20                        // padded LDS row stride (floats)
#define LTOT (NSLICE * NROW * LROW)    // 3600 floats = 14.4 KB
#define NLOADN (NSLICE * NROW * NCOL)  // 3240 elements to stage

__global__ __launch_bounds__(256) void coeff_to_value_kernel(
    const float* __restrict__ in, float* __restrict__ out, int nslab) {
  __shared__ float tile[LTOT];

  const int tid  = threadIdx.x;     // 0..255
  const int lane = tid & 31;
  const int wid  = tid >> 5;        // wave id 0..7 -> owns d = d0 + wid
  const int n    = lane & 15;       // N column (and A-matrix row M)
  const int hi   = lane >> 4;       // lane half: selects K sub-pair / M half

  const int w0   = blockIdx.x * TDW;
  const int h0   = blockIdx.y * TDH;
  const int dblk = blockIdx.z % (DD / DBLK);
  const int slab = blockIdx.z / (DD / DBLK);
  const int d0   = dblk * DBLK;
  if (slab >= nslab) return;  // uniform per block

  const size_t slabElems = (size_t)DD * HH * WW;
  const float* slabIn = in + (size_t)slab * slabElems;

  // Zero-fill LDS: implements bound='zero' halo for every OOB tap.
  for (int i = tid; i < LTOT; i += 256) tile[i] = 0.0f;
  __syncthreads();

  // Low 32 bits of a flat shared pointer == wave-relative LDS byte offset
  // (LDS aperture lives in addr[63:32], ISA 10.2).
  const unsigned ldsBase = (unsigned)(uintptr_t)&tile[0];
  const unsigned long long sbase = (unsigned long long)(uintptr_t)slabIn;

  // Stage haloed 10x18x18 region via CDNA5 async global->LDS copies (GVS
  // addressing: SGPR-pair base + per-lane i32 byte offset). EXEC-predicated
  // by the bounds check; OOB entries keep their LDS zeros.
  for (int i = tid; i < NLOADN; i += 256) {
    int dz  = i / (NROW * NCOL);
    int r   = i - dz * (NROW * NCOL);
    int row = r / NCOL;
    int col = r - row * NCOL;
    int gd = d0 - 1 + dz;
    int gh = h0 - 1 + row;
    int gw = w0 - 1 + col;
    if ((unsigned)gd < (unsigned)DD && (unsigned)gh < (unsigned)HH &&
        (unsigned)gw < (unsigned)WW) {
      unsigned voff = (unsigned)(((gd * HH + gh) * WW + gw) * 4);
      unsigned ldst = ldsBase + (unsigned)(((dz * NROW + row) * LROW + col) * 4);
      asm volatile("global_load_async_to_lds_b32 %0, %1, %2"
                   :
                   : "v"(ldst), "v"(voff), "s"(sbase)
                   : "memory");
    }
  }
  asm volatile("s_wait_asynccnt 0" ::: "memory");
  __syncthreads();

  // Separable weights: [1/6, 4/6, 1/6] per axis.
  const float Wc = 2.0f / 3.0f;
  const float We = 1.0f / 6.0f;
  const float wz[3] = {We, Wc, We};

  // X[j, n] = D- and W-filtered value at staged row j, output column n,
  // for this wave's d-slice window {wid, wid+1, wid+2}.
  auto xval = [&](int j) -> float {
    float s = 0.0f;
#pragma unroll
    for (int dz = 0; dz < 3; ++dz) {
      const float* r = &tile[((wid + dz) * NROW + j) * LROW + n];
      s += wz[dz] * (We * r[0] + Wc * r[1] + We * r[2]);
    }
    return s;
  };

  // H-axis pass as fp32 WMMA chain: Out(16x16) = FH(16x20 zero-padded) * X(20x16).
  // A (16x4 f32) layout: lane L holds M=L&15; VGPR0 K=(L<16?0:2)+4c, VGPR1 K+1.
  // B (4x16 f32) layout (A-symmetric): VGPR0 K=(L<16?0:2)+4c, N=L&15; VGPR1 K+1.
  const int kbase = hi * 2;
  const int m = n;  // A-matrix row for this lane
  v8f acc = {};
#pragma unroll
  for (int c = 0; c < 5; ++c) {
    const int K0 = c * 4 + kbase;
    const int K1 = K0 + 1;
    // FH band: FH[m, j] = {We, Wc, We} at j = m, m+1, m+2 (taps -1,0,+1).
    float a0 = (K0 == m) ? We : (K0 == m + 1) ? Wc : (K0 == m + 2) ? We : 0.0f;
    float a1 = (K1 == m) ? We : (K1 == m + 1) ? Wc : (K1 == m + 2) ? We : 0.0f;
    float b0 = (K0 < NROW) ? xval(K0) : 0.0f;  // K = 18,19 are zero padding
    float b1 = (K1 < NROW) ? xval(K1) : 0.0f;
    v2f A = {a0, a1};
    v2f B = {b0, b1};
    acc = __builtin_amdgcn_wmma_f32_16x16x4_f32(
        /*neg_a=*/false, A, /*neg_b=*/false, B,
        /*c_mod=*/(short)0, acc, /*reuse_a=*/false, /*reuse_b=*/false);
  }

  // D (16x16 f32) layout: VGPR r -> M = r + (lane<16 ? 0 : 8), N = lane&15.
  const int d = d0 + wid;
  size_t obase = (((size_t)slab * DD + d) * HH + (h0 + hi * 8)) * WW + (w0 + n);
#pragma unroll
  for (int r = 0; r < 8; ++r) {
    out[obase + (size_t)r * WW] = acc[r];
  }
}

extern "C" void kernel_launch(void* const* d_in, const int* in_sizes, int n_in,
                              void* d_out, int out_size, void* d_ws, size_t ws_size,
                              hipStream_t stream) {
  (void)n_in; (void)out_size; (void)d_ws; (void)ws_size;
  const float* in = (const float*)d_in[0];
  float* out = (float*)d_out;

  const int slabElems = DD * HH * WW;
  const int nslab = in_sizes[0] / slabElems;  // = 4 for (2,2,D,H,W)

  dim3 block(256, 1, 1);
  dim3 grid(WW / TDW, HH / TDH, (unsigned)((DD / DBLK) * nslab));
  hipLaunchKernelGGL(coeff_to_value_kernel, grid, block, 0, stream, in, out, nslab);
}